// CostVolume_51694226375015
// MI455X (gfx1250) — compile-verified
//
#include <hip/hip_runtime.h>

#define DEV __device__ __forceinline__

// Fixed problem sizes from setup_inputs(): V=3, b=1, c=32, h=128, w=160, D=128
constexpr int Vv = 3;
constexpr int Cc = 32;
constexpr int Hh = 128;
constexpr int Ww = 160;
constexpr int Dd = 128;
constexpr int HW = Hh * Ww;                   // 20480 elements / channel plane
constexpr unsigned PLANE_BYTES = HW * 4u;     // 81920 B
constexpr int TPB = 1024;                     // 32 wave32 waves
constexpr int YSPLIT = 2;
constexpr int PIX_PER_BLOCK = HW / YSPLIT;    // 10240
constexpr int PPT = PIX_PER_BLOCK / TPB;      // 10 pixels per thread

typedef unsigned int u32x4 __attribute__((ext_vector_type(4)));
typedef int i32x4 __attribute__((ext_vector_type(4)));
typedef int i32x8 __attribute__((ext_vector_type(8)));

// ---------------------------------------------------------------------------
// TDM: stage one contiguous 80KB channel plane (global -> LDS) via the
// Tensor Data Mover. Descriptor per CDNA5 ISA:
//   group0: count=1 | lds_addr | global_addr | type=2
//   group1: data_size=4B, tensor_dim0=tile_dim0=20480, dim1=1, strides=20480
// Groups 2/3 zero (unused higher dims: tile_dim3/4 = 0).
// ---------------------------------------------------------------------------
DEV void tdm_load_plane(const float* gsrc, unsigned lds_off) {
  unsigned long long ga = (unsigned long long)(const void*)gsrc;
  u32x4 g0;
  g0[0] = 1u;                                        // count=1, user mode
  g0[1] = lds_off;                                   // lds_addr (bytes)
  g0[2] = (unsigned)(ga & 0xffffffffull);            // global_addr[31:0]
  g0[3] = (unsigned)((ga >> 32) & 0x1ffffffull)      // global_addr[56:32]
        | (2u << 30);                                // type=2 ("image")
  i32x8 g1;
  g1[0] = 0x00020000;                                // wg_mask=0, data_size=2 (4B)
  g1[1] = (int)((HW & 0xffff) << 16);                // tensor_dim0[15:0] @ [63:48]
  g1[2] = (int)((((unsigned)HW >> 16) & 0xffffu))    // tensor_dim0[31:16]
        | (1 << 16);                                 // tensor_dim1[15:0] = 1
  g1[3] = (int)((HW & 0xffff) << 16);                // tile_dim0 = 20480
  g1[4] = 1;                                         // tile_dim1=1, tile_dim2=0
  g1[5] = (int)HW;                                   // tensor_dim0_stride[31:0]
  g1[6] = (int)((HW & 0xffff) << 16);                // stride0 hi=0 | stride1 lo16
  g1[7] = 0;                                         // stride1 hi
  i32x4 z4 = {0, 0, 0, 0};
#if defined(__clang_major__) && (__clang_major__ >= 23)
  i32x8 z8 = {0, 0, 0, 0, 0, 0, 0, 0};
  __builtin_amdgcn_tensor_load_to_lds(g0, g1, z4, z4, z8, 0);
#else
  __builtin_amdgcn_tensor_load_to_lds(g0, g1, z4, z4, 0);
#endif
}

DEV int iclamp(int v, int lo, int hi) { return v < lo ? lo : (v > hi ? hi : v); }

// Bilinear sample from an LDS-resident channel plane, replicating the
// reference's per-tap validity (pre-clamp bounds test, clamped index).
DEV float bilerp_lds(const float* __restrict__ pl, float px, float py) {
  float fx0 = floorf(px);
  float fy0 = floorf(py);
  float wx1 = px - fx0;
  float wy1 = py - fy0;
  float wx0 = 1.0f - wx1;
  float wy0 = 1.0f - wy1;
  int x0 = (int)fx0, y0 = (int)fy0;
  int x1 = x0 + 1,   y1 = y0 + 1;
  float vx0 = (x0 >= 0 && x0 <= Ww - 1) ? 1.0f : 0.0f;
  float vx1 = (x1 >= 0 && x1 <= Ww - 1) ? 1.0f : 0.0f;
  float vy0 = (y0 >= 0 && y0 <= Hh - 1) ? 1.0f : 0.0f;
  float vy1 = (y1 >= 0 && y1 <= Hh - 1) ? 1.0f : 0.0f;
  int xc0 = iclamp(x0, 0, Ww - 1), xc1 = iclamp(x1, 0, Ww - 1);
  int yc0 = iclamp(y0, 0, Hh - 1), yc1 = iclamp(y1, 0, Hh - 1);
  float v00 = pl[yc0 * Ww + xc0] * (vx0 * vy0);
  float v10 = pl[yc0 * Ww + xc1] * (vx1 * vy0);
  float v01 = pl[yc1 * Ww + xc0] * (vx0 * vy1);
  float v11 = pl[yc1 * Ww + xc1] * (vx1 * vy1);
  return v00 * (wx0 * wy0) + v10 * (wx1 * wy0) +
         v01 * (wx0 * wy1) + v11 * (wx1 * wy1);
}

DEV void inv3(const float* a, float* o) {
  float c00 = a[4] * a[8] - a[5] * a[7];
  float c01 = a[5] * a[6] - a[3] * a[8];
  float c02 = a[3] * a[7] - a[4] * a[6];
  float det = a[0] * c00 + a[1] * c01 + a[2] * c02;
  float id = 1.0f / det;
  o[0] = c00 * id;
  o[1] = (a[2] * a[7] - a[1] * a[8]) * id;
  o[2] = (a[1] * a[5] - a[2] * a[4]) * id;
  o[3] = c01 * id;
  o[4] = (a[0] * a[8] - a[2] * a[6]) * id;
  o[5] = (a[2] * a[3] - a[0] * a[5]) * id;
  o[6] = c02 * id;
  o[7] = (a[1] * a[6] - a[0] * a[7]) * id;
  o[8] = (a[0] * a[4] - a[1] * a[3]) * id;
}

// ---------------------------------------------------------------------------
// Kernel A: per source view v in {1,2} compute M = R_rel * inv(K0), t = t_rel
// where rel = E_v * inv(E0) (rigid inverse). 24 floats -> d_ws.
// ---------------------------------------------------------------------------
__global__ void setup_xf_kernel(const float* __restrict__ K,
                                const float* __restrict__ E,
                                float* __restrict__ xf) {
  if (threadIdx.x != 0 || blockIdx.x != 0) return;
  float iK[9];
  inv3(K, iK);  // view 0 intrinsics, b=0
  const float* E0 = E;
  float R0t[9] = {E0[0], E0[4], E0[8],
                  E0[1], E0[5], E0[9],
                  E0[2], E0[6], E0[10]};
  float t0[3] = {E0[3], E0[7], E0[11]};
  for (int v = 1; v < Vv; ++v) {
    const float* Ev = E + v * 16;
    float Rr[9], tr[3];
    for (int i = 0; i < 3; ++i)
      for (int j = 0; j < 3; ++j) {
        float s = 0.f;
        for (int k = 0; k < 3; ++k) s += Ev[i * 4 + k] * R0t[k * 3 + j];
        Rr[i * 3 + j] = s;
      }
    for (int i = 0; i < 3; ++i)
      tr[i] = Ev[i * 4 + 3] -
              (Rr[i * 3 + 0] * t0[0] + Rr[i * 3 + 1] * t0[1] + Rr[i * 3 + 2] * t0[2]);
    float* o = xf + (v - 1) * 12;
    for (int i = 0; i < 3; ++i)
      for (int j = 0; j < 3; ++j) {
        float s = 0.f;
        for (int k = 0; k < 3; ++k) s += Rr[i * 3 + k] * iK[k * 3 + j];
        o[i * 3 + j] = s;
      }
    o[9] = tr[0]; o[10] = tr[1]; o[11] = tr[2];
  }
}

// ---------------------------------------------------------------------------
// Kernel B: one block per (depth slice d, y-half). 1024 threads / 32 waves.
// Source channel planes TDM-staged into LDS, double buffered (4 x 80KB = 320KB).
// ---------------------------------------------------------------------------
__global__ __launch_bounds__(TPB) void cost_volume_kernel(
    const float* __restrict__ feat, const float* __restrict__ depth_values,
    const float* __restrict__ xf, float* __restrict__ out) {
  extern __shared__ char smem[];

  const int d = blockIdx.x;
  const int half = blockIdx.y;
  const int tid = (int)threadIdx.x;
  const float depth = depth_values[d];

  // Uniform transforms (9 + 3 floats per source view).
  float M1[9], T1[3], M2[9], T2[3];
#pragma unroll
  for (int i = 0; i < 9; ++i) { M1[i] = xf[i]; M2[i] = xf[12 + i]; }
#pragma unroll
  for (int i = 0; i < 3; ++i) { T1[i] = xf[9 + i]; T2[i] = xf[21 + i]; }

  constexpr float SX = (float)Ww / (float)(Ww - 1);  // 160/159
  constexpr float SY = (float)Hh / (float)(Hh - 1);  // 128/127

  // Precompute projected coords once per pixel; channel loop reuses them.
  float px1[PPT], py1[PPT], px2[PPT], py2[PPT];
  const int pbase = half * PIX_PER_BLOCK + tid;
#pragma unroll
  for (int k = 0; k < PPT; ++k) {
    int p = pbase + k * TPB;
    float fx = (float)(p % Ww);
    float fy = (float)(p / Ww);
    {
      float X = (M1[0] * fx + M1[1] * fy + M1[2]) * depth + T1[0];
      float Y = (M1[3] * fx + M1[4] * fy + M1[5]) * depth + T1[1];
      float Z = (M1[6] * fx + M1[7] * fy + M1[8]) * depth + T1[2];
      float iz = 1.0f / Z;
      px1[k] = X * iz * SX - 0.5f;
      py1[k] = Y * iz * SY - 0.5f;
    }
    {
      float X = (M2[0] * fx + M2[1] * fy + M2[2]) * depth + T2[0];
      float Y = (M2[3] * fx + M2[4] * fy + M2[5]) * depth + T2[1];
      float Z = (M2[6] * fx + M2[7] * fy + M2[8]) * depth + T2[2];
      float iz = 1.0f / Z;
      px2[k] = X * iz * SX - 0.5f;
      py2[k] = Y * iz * SY - 0.5f;
    }
  }

  const unsigned lds_base = (unsigned)(size_t)smem;  // low 32 bits = LDS offset

  // Prologue: stage channel 0 (views 1 and 2) into buffer 0.
  if (tid < 32) {
    tdm_load_plane(feat + (size_t)(1 * Cc + 0) * HW, lds_base);
    tdm_load_plane(feat + (size_t)(2 * Cc + 0) * HW, lds_base + PLANE_BYTES);
  }

  for (int c = 0; c < Cc; ++c) {
    if (tid < 32) {
      if (c + 1 < Cc) {
        unsigned boff = lds_base + (unsigned)((c + 1) & 1) * (2u * PLANE_BYTES);
        tdm_load_plane(feat + (size_t)(1 * Cc + (c + 1)) * HW, boff);
        tdm_load_plane(feat + (size_t)(2 * Cc + (c + 1)) * HW, boff + PLANE_BYTES);
        __builtin_amdgcn_s_wait_tensorcnt(2);  // channel c's pair complete
      } else {
        __builtin_amdgcn_s_wait_tensorcnt(0);
      }
    }
    __syncthreads();  // buffer (c&1) holds channel c of views 1,2

    const float* p1 = (const float*)(smem + (size_t)(c & 1) * (2u * PLANE_BYTES));
    const float* p2 = p1 + HW;
    const float* ref_plane = feat + (size_t)c * HW;          // view 0
    float* out_plane = out + ((size_t)c * Dd + d) * HW;

#pragma unroll
    for (int k = 0; k < PPT; ++k) {
      int p = pbase + k * TPB;
      float r  = ref_plane[p];
      float s1 = bilerp_lds(p1, px1[k], py1[k]);
      float s2 = bilerp_lds(p2, px2[k], py2[k]);
      float sum = r + s1 + s2;
      float sq  = r * r + s1 * s1 + s2 * s2;
      float m = sum * (1.0f / 3.0f);
      float res = sq * (1.0f / 3.0f) - m * m;
      __builtin_nontemporal_store(res, out_plane + p);  // streaming 335MB output
    }
    __syncthreads();  // all reads of buffer (c&1) done before it is re-staged
  }
}

extern "C" void kernel_launch(void* const* d_in, const int* in_sizes, int n_in,
                              void* d_out, int out_size, void* d_ws, size_t ws_size,
                              hipStream_t stream) {
  const float* depth = (const float*)d_in[0];  // (1,128)
  const float* feat  = (const float*)d_in[1];  // (3,1,32,128,160)
  const float* K     = (const float*)d_in[2];  // (3,1,3,3)
  const float* E     = (const float*)d_in[3];  // (3,1,4,4)
  float* out = (float*)d_out;                  // (1,32,128,128,160)
  float* xf  = (float*)d_ws;                   // 24 floats

  setup_xf_kernel<<<dim3(1), dim3(32), 0, stream>>>(K, E, xf);

  dim3 grid(Dd, YSPLIT);
  size_t shmem = 4ull * PLANE_BYTES;  // 327680 B: double-buffered 2-view planes
  cost_volume_kernel<<<grid, dim3(TPB), shmem, stream>>>(feat, depth, xf, out);
}